// CombinedMetricDiffCE_19456201851259
// MI455X (gfx1250) — compile-verified
//
#include <hip/hip_runtime.h>
#include <cstdint>

// ---------------------------------------------------------------------------
// CombinedMetricDiffCE for MI455X (gfx1250)
//   out = 0.5*mean(LABEL_W[argmax(x), t]^2) + 0.5*mean(label_smooth_CE(x, t))
// B = 2,000,000 rows, C = 26 classes, fp32.  Pure HBM-bandwidth problem
// (~224 MB -> ~9.6 us floor at 23.3 TB/s).  Data movement uses the CDNA5
// async global->LDS path (GLOBAL_LOAD_ASYNC_TO_LDS_B128 + s_wait_asynccnt),
// double-buffered per block so DMA of tile i+1 overlaps math on tile i.
// The 26x26 squared cosine-distance table is fully evaluated at COMPILE TIME
// (constexpr, f64 norms / f32 dot exactly like the numpy reference) and only
// copied const->LDS at kernel start.
// ---------------------------------------------------------------------------

#define NCLASS     26
#define TILE_ROWS  256
#define ROW_BYTES  (NCLASS * 4)             // 104 bytes per row
#define TILE_BYTES (TILE_ROWS * ROW_BYTES)  // 26624 bytes per tile
#define BLOCK      256                      // 8 waves (wave32)
#define MAX_BLOCKS 2048

// label-smoothing constants: eps = 0.1/26
//   w_on - w_off = 1 - 0.1/25 = 0.996 exactly; w_off = 0.1/650
#define CE_DIFF 0.996f
#define CE_WOFF 1.5384615384615385e-04f

// ---------------- compile-time LABEL_W**2 table ----------------------------
#define SV 0.7071f
constexpr float k_dirs[NCLASS][3] = {
    {0.f, 0.f, 1.f},   {0.f, 0.f, -1.f},  {0.f, -SV, SV},    {0.f, -1.f, 0.f},
    {0.f, -SV, -SV},   {0.f, SV, -SV},    {0.f, 1.f, 0.f},   {0.f, SV, SV},
    {SV, 0.f, SV},     {1.f, 0.f, 0.f},   {SV, 0.f, -SV},    {-SV, 0.f, -SV},
    {-1.f, 0.f, 0.f},  {-SV, 0.f, SV},    {0.5f, -SV, 0.5f}, {-0.5f, -SV, -0.5f},
    {-0.5f, SV, -0.5f},{0.5f, SV, 0.5f},  {SV, -SV, 0.f},    {-SV, -SV, 0.f},
    {-SV, SV, 0.f},    {SV, SV, 0.f},     {0.5f, -SV, -0.5f},{-0.5f, -SV, 0.5f},
    {-0.5f, SV, 0.5f}, {0.5f, SV, -0.5f}};

constexpr double csqrt(double x) {
  double g = (x > 1.0) ? x : 1.0;
  for (int i = 0; i < 64; ++i) g = 0.5 * (g + x / g);
  return g;
}

struct W2Table { float v[NCLASS * NCLASS]; };

constexpr W2Table make_w2() {
  W2Table t{};
  for (int i = 0; i < NCLASS; ++i) {
    for (int j = 0; j < NCLASS; ++j) {
      // f32 dot (numpy f32 matmul), f64 norms (np.linalg.norm), f64 divide,
      // (1 - cos) cast to f32, squared in f32 -- mirrors the reference.
      const float dotf = k_dirs[i][0] * k_dirs[j][0] + k_dirs[i][1] * k_dirs[j][1] +
                         k_dirs[i][2] * k_dirs[j][2];
      double ni = csqrt((double)k_dirs[i][0] * k_dirs[i][0] +
                        (double)k_dirs[i][1] * k_dirs[i][1] +
                        (double)k_dirs[i][2] * k_dirs[i][2]);
      double nj = csqrt((double)k_dirs[j][0] * k_dirs[j][0] +
                        (double)k_dirs[j][1] * k_dirs[j][1] +
                        (double)k_dirs[j][2] * k_dirs[j][2]);
      ni = (ni > 1e-8) ? ni : 1e-8;
      nj = (nj > 1e-8) ? nj : 1e-8;
      const float w = (float)(1.0 - (double)dotf / (ni * nj));
      t.v[i * NCLASS + j] = w * w;
    }
  }
  return t;
}

__device__ __constant__ W2Table c_w2 = make_w2();

// ---------------- async DMA helpers ----------------------------------------
// one lane: copy 16 bytes global -> LDS via the async data mover (ASYNCcnt)
__device__ __forceinline__ void async_ld16(uint32_t lds_off, uint64_t gaddr) {
  asm volatile("global_load_async_to_lds_b128 %0, %1, off"
               :
               : "v"(lds_off), "v"(gaddr)
               : "memory");
}

// Stage one 256-row tile into LDS.  Every wave issues EXACTLY 7 async b128
// instructions (indices clamped into range, all lanes active) so that
// "s_wait_asynccnt 7" is a precise per-wave double-buffer threshold.
// NOTE: rows*104 is a multiple of 16 for every tile here (B=2e6 -> last tile
// has 128 rows), so b128 chunks cover the tile exactly with no OOB reads.
__device__ __forceinline__ void stage_tile(uint64_t xg, uint32_t lds_base,
                                           int tile, int B) {
  int rows = B - tile * TILE_ROWS;
  if (rows > TILE_ROWS) rows = TILE_ROWS;
  const int cmax = (rows * ROW_BYTES) / 16 - 1;
  const uint64_t tbase = xg + (uint64_t)tile * (uint64_t)TILE_BYTES;
#pragma unroll
  for (int it = 0; it < 7; ++it) {
    int c = it * BLOCK + (int)threadIdx.x;
    if (c > cmax) c = cmax;  // duplicate copies are benign (same src -> same dst)
    async_ld16(lds_base + (uint32_t)(c * 16), tbase + (uint64_t)(c * 16));
  }
}

// ---------------- main kernel ----------------------------------------------
__global__ __launch_bounds__(BLOCK) void cmce_main_kernel(
    const float* __restrict__ x, const uint32_t* __restrict__ tgt2 /*int64 lo/hi*/,
    double* __restrict__ partials, int B, int ntiles) {
  __shared__ __align__(16) char tb[2][TILE_BYTES];  // double-buffered tiles
  __shared__ float w2[NCLASS * NCLASS];             // LABEL_W**2 table
  __shared__ float reds[BLOCK];

  const int tid = (int)threadIdx.x;

  const uint64_t xg = (uint64_t)(uintptr_t)x;
  // low 32 bits of a generic pointer into LDS == wave-relative LDS byte offset
  const uint32_t lds0 = (uint32_t)(uintptr_t)&tb[0][0];
  const uint32_t lds1 = (uint32_t)(uintptr_t)&tb[1][0];

  int tile = (int)blockIdx.x;
  const int stride = (int)gridDim.x;
  int k = 0;

  if (tile < ntiles) stage_tile(xg, lds0, tile, B);  // prologue: tile0 -> buf0

  // copy the precomputed table to LDS (overlaps with the prologue DMA)
  for (int idx = tid; idx < NCLASS * NCLASS; idx += BLOCK) w2[idx] = c_w2.v[idx];

  float acc_dir = 0.0f;
  float acc_ce = 0.0f;

  for (; tile < ntiles; tile += stride, k ^= 1) {
    const int next = tile + stride;  // block-uniform
    if (next < ntiles) {
      // prefetch next tile into the other buffer (its readers finished at the
      // end-of-iteration barrier of the previous loop trip)
      stage_tile(xg, k ? lds0 : lds1, next, B);
      // wait for the CURRENT tile's 7 async loads; the 7 just-issued
      // prefetch loads may remain outstanding (in-order completion).
      asm volatile("s_wait_asynccnt 7" ::: "memory");
    } else {
      asm volatile("s_wait_asynccnt 0" ::: "memory");
    }
    __syncthreads();  // publish staged tile across waves (and w2 on trip 0)

    const int r = tile * TILE_ROWS + tid;  // one row per thread per tile
    if (r < B) {
      const float* row = (const float*)&tb[k][tid * ROW_BYTES];
      float v[NCLASS];
#pragma unroll
      for (int i = 0; i < 13; ++i) {  // 13 x ds_load_b64, bank-conflict-free
        const float2 t2 = ((const float2*)row)[i];
        v[2 * i] = t2.x;
        v[2 * i + 1] = t2.y;
      }
      // max + first-occurrence argmax
      float m = v[0];
      int p = 0;
#pragma unroll
      for (int i = 1; i < NCLASS; ++i) {
        if (v[i] > m) { m = v[i]; p = i; }
      }
      float se = 0.0f, sx = 0.0f;
#pragma unroll
      for (int i = 0; i < NCLASS; ++i) {
        se += __expf(v[i] - m);
        sx += v[i];
      }
      const float lse = m + __logf(se);
      const int t = (int)tgt2[(size_t)2 * (size_t)r];  // int64 target, low dword
      const float xt = row[t];                          // dynamic-index ds_load_b32
      const float ce = -(CE_DIFF * (xt - lse) + CE_WOFF * (sx - (float)NCLASS * lse));
      acc_dir += w2[p * NCLASS + t];
      acc_ce += ce;
    }
    __syncthreads();  // all reads of tb[k] done before it is re-staged next trip
  }

  // ---- deterministic block reduction -> per-block f64 partials ----
  reds[tid] = acc_dir;
  __syncthreads();
#pragma unroll
  for (int s = BLOCK / 2; s > 0; s >>= 1) {
    if (tid < s) reds[tid] += reds[tid + s];
    __syncthreads();
  }
  if (tid == 0) partials[blockIdx.x] = (double)reds[0];
  __syncthreads();

  reds[tid] = acc_ce;
  __syncthreads();
#pragma unroll
  for (int s = BLOCK / 2; s > 0; s >>= 1) {
    if (tid < s) reds[tid] += reds[tid + s];
    __syncthreads();
  }
  if (tid == 0) partials[gridDim.x + blockIdx.x] = (double)reds[0];
}

// ---------------- deterministic finalize -----------------------------------
__global__ __launch_bounds__(256) void cmce_finalize_kernel(
    const double* __restrict__ partials, int nblk, int B, float* __restrict__ out) {
  __shared__ double sd[256];
  __shared__ double sc[256];
  const int tid = (int)threadIdx.x;
  double a = 0.0, b = 0.0;
  for (int i = tid; i < nblk; i += 256) {
    a += partials[i];
    b += partials[nblk + i];
  }
  sd[tid] = a;
  sc[tid] = b;
  __syncthreads();
#pragma unroll
  for (int s = 128; s > 0; s >>= 1) {
    if (tid < s) {
      sd[tid] += sd[tid + s];
      sc[tid] += sc[tid + s];
    }
    __syncthreads();
  }
  if (tid == 0) {
    const double invB = 1.0 / (double)B;
    out[0] = (float)(0.5 * sd[0] * invB + 0.5 * sc[0] * invB);
  }
}

extern "C" void kernel_launch(void* const* d_in, const int* in_sizes, int n_in,
                              void* d_out, int out_size, void* d_ws, size_t ws_size,
                              hipStream_t stream) {
  (void)n_in;
  (void)out_size;
  (void)ws_size;
  const float* x = (const float*)d_in[0];
  const uint32_t* tgt2 = (const uint32_t*)d_in[1];  // int64 viewed as dword pairs
  const int B = in_sizes[1];                        // number of rows / targets
  const int ntiles = (B + TILE_ROWS - 1) / TILE_ROWS;
  int nblk = MAX_BLOCKS;
  if (nblk > ntiles) nblk = ntiles;

  double* partials = (double*)d_ws;  // [2 * nblk] doubles, rewritten every call

  cmce_main_kernel<<<nblk, BLOCK, 0, stream>>>(x, tgt2, partials, B, ntiles);
  cmce_finalize_kernel<<<1, 256, 0, stream>>>(partials, nblk, B, (float*)d_out);
}